// get_model_6047313953116
// MI455X (gfx1250) — compile-verified
//
#include <hip/hip_runtime.h>

typedef __attribute__((ext_vector_type(2))) float v2f;
typedef __attribute__((ext_vector_type(8))) float v8f;

#define BIGF 3.0e38f

// ---------------------------------------------------------------------------
// Prep: A_packed[i] = (-2*wx, -2*wy, -2*wz, |w|^2), w = xyz_ref + flow
// ---------------------------------------------------------------------------
__global__ void prep_pack_kernel(const float* __restrict__ xyz_ref,
                                 const float* __restrict__ flow,
                                 float4* __restrict__ apack, int n1) {
    int i = blockIdx.x * blockDim.x + threadIdx.x;
    if (i >= n1) return;
    float wx = xyz_ref[3 * i + 0] + flow[3 * i + 0];
    float wy = xyz_ref[3 * i + 1] + flow[3 * i + 1];
    float wz = xyz_ref[3 * i + 2] + flow[3 * i + 2];
    float4 a;
    a.x = -2.0f * wx;
    a.y = -2.0f * wy;
    a.z = -2.0f * wz;
    a.w = wx * wx + wy * wy + wz * wz;
    apack[i] = a;
}

// Branchless sorted-top3 insert (d0 <= d1 <= d2)
__device__ __forceinline__ void insert3(float x, int xi,
                                        float& d0, float& d1, float& d2,
                                        int& i0, int& i1, int& i2) {
    bool c0 = x < d0, c1 = x < d1, c2 = x < d2;
    d2 = c1 ? d1 : (c2 ? x : d2);
    i2 = c1 ? i1 : (c2 ? xi : i2);
    d1 = c0 ? d0 : (c1 ? x : d1);
    i1 = c0 ? i0 : (c1 ? xi : i1);
    d0 = c0 ? x : d0;
    i0 = c0 ? xi : i0;
}

// ---------------------------------------------------------------------------
// Main: each wave (32 lanes, one per block) owns 16 queries.
// Distance tiles via V_WMMA_F32_16X16X4_F32; lane = query column.
// ---------------------------------------------------------------------------
__global__ __launch_bounds__(32)
void knn_warp_kernel(const float* __restrict__ xyz_cur,
                     const float* __restrict__ feat,
                     const float4* __restrict__ apack,
                     float* __restrict__ out,
                     int n1, int C) {
    const int lane = threadIdx.x;            // 0..31 (wave32)
    const int qbase = blockIdx.x * 16;
    const int qidx = qbase + (lane & 15);
    const bool hi = lane >= 16;

    // Query for this lane (lane pair l, l+16 share a query).
    float qx = xyz_cur[3 * qidx + 0];
    float qy = xyz_cur[3 * qidx + 1];
    float qz = xyz_cur[3 * qidx + 2];
    float qn = qx * qx + qy * qy + qz * qz;

    // B matrix (4x16): lanes 0-15 hold rows K=0,1; lanes 16-31 rows K=2,3.
    v2f b;
    b.x = hi ? qz : qx;
    b.y = hi ? 1.0f : qy;

    float d0 = BIGF, d1 = BIGF, d2v = BIGF;
    int i0 = 0, i1 = 0, i2 = 0;

    for (int t = 0; t < n1; t += 16) {
        // A matrix (16x4): lane m (0-15) holds K=0,1; lane m+16 holds K=2,3.
        float4 ap = apack[t + (lane & 15)];
        v2f a;
        a.x = hi ? ap.z : ap.x;
        a.y = hi ? ap.w : ap.y;

        v8f acc = {};
        // D[m][n] = -2 * p_m . q_n + |p_m|^2   (|q|^2 dropped: rank-invariant)
        acc = __builtin_amdgcn_wmma_f32_16x16x4_f32(
            false, a, false, b, (short)0, acc, false, false);

        // Wave-uniform early-out: does ANY lane have a candidate beating its
        // current 3rd-best? (min-tree folds to v_min3; vote keeps the branch
        // uniform so EXEC remains all-ones for the WMMA.)
        float m01 = fminf(acc[0], acc[1]);
        float m23 = fminf(acc[2], acc[3]);
        float m45 = fminf(acc[4], acc[5]);
        float m67 = fminf(acc[6], acc[7]);
        float mn = fminf(fminf(m01, m23), fminf(m45, m67));

        if (__any(mn < d2v)) {
            // Lane (n, lo) holds rows m=0..7; lane (n, hi) holds m=8..15.
            const int pb = t + (hi ? 8 : 0);
#pragma unroll
            for (int v = 0; v < 8; ++v) {
                insert3(acc[v], pb + v, d0, d1, d2v, i0, i1, i2);
            }
        }
    }

    // Merge the two half-tiles of each query across the lane pair (l <-> l^16).
    {
        float od0 = __shfl_xor(d0, 16, 32);
        float od1 = __shfl_xor(d1, 16, 32);
        float od2 = __shfl_xor(d2v, 16, 32);
        int oi0 = __shfl_xor(i0, 16, 32);
        int oi1 = __shfl_xor(i1, 16, 32);
        int oi2 = __shfl_xor(i2, 16, 32);
        insert3(od0, oi0, d0, d1, d2v, i0, i1, i2);
        insert3(od1, oi1, d0, d1, d2v, i0, i1, i2);
        insert3(od2, oi2, d0, d1, d2v, i0, i1, i2);
    }

    // Inverse-distance weights. True d^2 = D + |q|^2 (clamped), dist >= 1e-10.
    float s0 = sqrtf(fmaxf(d0 + qn, 0.0f));
    float s1 = sqrtf(fmaxf(d1 + qn, 0.0f));
    float s2 = sqrtf(fmaxf(d2v + qn, 0.0f));
    float inv0 = 1.0f / fmaxf(s0, 1e-10f);
    float inv1 = 1.0f / fmaxf(s1, 1e-10f);
    float inv2 = 1.0f / fmaxf(s2, 1e-10f);
    float rs = 1.0f / (inv0 + inv1 + inv2);
    float w0 = inv0 * rs, w1 = inv1 * rs, w2 = inv2 * rs;

    // Feature gather: broadcast each query's (idx, w) to the whole wave,
    // lanes stride the C channels (coalesced row reads of feat).
    for (int q = 0; q < 16; ++q) {
        int bi0 = __shfl(i0, q, 32);
        int bi1 = __shfl(i1, q, 32);
        int bi2 = __shfl(i2, q, 32);
        float bw0 = __shfl(w0, q, 32);
        float bw1 = __shfl(w1, q, 32);
        float bw2 = __shfl(w2, q, 32);
        const float* f0 = feat + (size_t)bi0 * C;
        const float* f1 = feat + (size_t)bi1 * C;
        const float* f2 = feat + (size_t)bi2 * C;
        float* orow = out + (size_t)(qbase + q) * C;
        for (int c = lane; c < C; c += 32) {
            orow[c] = bw0 * f0[c] + bw1 * f1[c] + bw2 * f2[c];
        }
    }
}

// ---------------------------------------------------------------------------
// Launch
// ---------------------------------------------------------------------------
extern "C" void kernel_launch(void* const* d_in, const int* in_sizes, int n_in,
                              void* d_out, int out_size, void* d_ws, size_t ws_size,
                              hipStream_t stream) {
    const float* xyz_ref = (const float*)d_in[0];
    const float* flow    = (const float*)d_in[1];
    const float* xyz_cur = (const float*)d_in[2];
    const float* feat    = (const float*)d_in[3];
    float* out = (float*)d_out;

    const int n1 = in_sizes[0] / 3;          // 16384 warped points
    const int n2 = in_sizes[2] / 3;          // 16384 queries
    const int C  = in_sizes[3] / n1;         // 64 channels

    float4* apack = (float4*)d_ws;           // n1 * 16 bytes of scratch

    prep_pack_kernel<<<(n1 + 255) / 256, 256, 0, stream>>>(xyz_ref, flow, apack, n1);
    knn_warp_kernel<<<n2 / 16, 32, 0, stream>>>(xyz_cur, feat, apack, out, n1, C);
}